// NCC_58368605553289
// MI455X (gfx1250) — compile-verified
//
#include <hip/hip_runtime.h>
#include <math.h>

// Problem geometry from the reference's setup_inputs()
#define HH     160
#define WW     192
#define NVIEW  4
#define HW     (HH*WW)
#define THRESH 0.5f

typedef float v2f __attribute__((ext_vector_type(2)));
typedef float v8f __attribute__((ext_vector_type(8)));

// d_out layout (reference return order, flattened):
//   [0,HW)      1 - prob
//   [HW,2HW)    depth_out
//   [2HW,5HW)   norm_out  ([3,H,W])
//   [5HW,6HW)   mask_out  (0/1 as float)
__global__ __launch_bounds__(256) void ncc_main(
    const float* __restrict__ src, const float* __restrict__ dst,
    const float* __restrict__ Ks,  const float* __restrict__ Rs,
    const float* __restrict__ Ts,  const float* __restrict__ depth,
    const float* __restrict__ norm, const int* __restrict__ dilp,
    float* __restrict__ out)
{
  // Lane-major per-thread LDS arrays: element e of thread t at [e*256 + t]
  __shared__ float scL[25*256];   // mean-centered src patch taps
  __shared__ float dpL[ 9*256];   // permuted dilated depth patch
  __shared__ float npL[27*256];   // permuted normal patch (k*3+c)
  __shared__ float cbL[84];       // PP[36] | PT[12] | Kinv[36]
  __shared__ float sD[256];       // scratch for WMMA 16x16 result

  const int tid = threadIdx.x;
  const int pix = blockIdx.x * 256 + tid;

  // ---- wave 0: PP_v = K_v * R_v for the 4 views, one V_WMMA_F32_16X16X4_F32 ----
  // A (16x4): row m=(v*3+i) holds K_v[i][0..2] (rows 12..15 zero, K=3 zero)
  // B (4x16): col n=(v*3+j) holds R_v[0..2][j]
  // D (16x16): PP_v[i][j] = D[v*3+i][v*3+j]  (block diagonal)
  if (tid < 32) {
    int m = tid & 15;
    int kbase = (tid >> 4) * 2;              // lanes 0-15: K=0,1; lanes 16-31: K=2,3
    v2f a, b;
    #pragma unroll
    for (int t = 0; t < 2; ++t) {
      int k = kbase + t;
      float av = 0.f, bv = 0.f;
      if (m < 12 && k < 3) {
        int v = m / 3, r = m % 3;            // r = K-row (for A) / R-col (for B)
        av = Ks[(v*2 + 1)*9 + r*3 + k];      // Ks[0][v][1][r][k]
        bv = Rs[v*9 + k*3 + r];              // Rs[0][v][k][r]
      }
      if (t == 0) { a.x = av; b.x = bv; } else { a.y = av; b.y = bv; }
    }
    v8f c = {};
    v8f dacc = __builtin_amdgcn_wmma_f32_16x16x4_f32(
        /*neg_a=*/false, a, /*neg_b=*/false, b,
        /*c_mod=*/(short)0, c, /*reuse_a=*/false, /*reuse_b=*/false);
    int mhigh = (tid >> 4) * 8;              // lanes 16-31 hold rows M=8..15
    #pragma unroll
    for (int r = 0; r < 8; ++r)
      sD[(r + mhigh)*16 + m] = dacc[r];
  }
  __syncthreads();
  if (tid < 84) {
    float val;
    if (tid < 36) {                          // PP
      int v = tid/9, i = (tid%9)/3, j = tid%3;
      val = sD[(v*3 + i)*16 + (v*3 + j)];
    } else if (tid < 48) {                   // PT = PP * T
      int q = tid - 36; int v = q/3, i = q%3;
      float s = 0.f;
      for (int j = 0; j < 3; ++j) s += sD[(v*3 + i)*16 + (v*3 + j)] * Ts[v*3 + j];
      val = s;
    } else {                                 // Kinv = Ks[0][v][0]
      int q = tid - 48; int v = q/9, r = q%9;
      val = Ks[(v*2 + 0)*9 + r];
    }
    cbL[tid] = val;
  }
  __syncthreads();

  const bool active = pix < HW;
  const int x = pix % WW, y = pix / WW;
  const int dil = dilp[0];

  float scm = 0.f, svar = 0.f;
  if (active) {
    // src 5x5 patch (zero pad), then center
    float smean = 0.f;
    #pragma unroll
    for (int t = 0; t < 25; ++t) {
      int py = y + t/5 - 2, px = x + t%5 - 2;
      float v = (px >= 0 && px < WW && py >= 0 && py < HH) ? src[py*WW + px] : 0.f;
      scL[t*256 + tid] = v;
      smean += v;
    }
    smean *= (1.f/25.f);
    #pragma unroll
    for (int t = 0; t < 25; ++t) {
      float s = scL[t*256 + tid] - smean;
      scL[t*256 + tid] = s;
      scm  += s;
      svar += s*s;
    }
    scm  *= (1.f/25.f);
    svar *= (1.f/25.f);

    // depth (dilated) + normal patches, with perm swapping taps 0 and 4
    const int perm[9] = {4,1,2,3,0,5,6,7,8};
    #pragma unroll
    for (int k = 0; k < 9; ++k) {
      int sI = perm[k];
      int dyI = sI/3 - 1, dxI = sI%3 - 1;
      int pyd = y + dyI*dil, pxd = x + dxI*dil;
      dpL[k*256 + tid] = (pxd >= 0 && pxd < WW && pyd >= 0 && pyd < HH)
                         ? depth[pyd*WW + pxd] : 0.f;
      int py1 = y + dyI, px1 = x + dxI;
      bool in1 = (px1 >= 0 && px1 < WW && py1 >= 0 && py1 < HH);
      #pragma unroll
      for (int cch = 0; cch < 3; ++cch)
        npL[(k*3 + cch)*256 + tid] = in1 ? norm[cch*HW + py1*WW + px1] : 0.f;
    }
  }
  if (!active) return;   // no barriers below; each thread touches only its own LDS lanes

  float best = -1e30f; int bidx = 0; int bmany = 0;
  for (int d = 0; d < 9; ++d) {
    float dd  = dpL[d*256 + tid];
    float ndx = npL[(d*3 + 0)*256 + tid];
    float ndy = npL[(d*3 + 1)*256 + tid];
    float ndz = npL[(d*3 + 2)*256 + tid];
    float sco; int manyd = 0;
    if (dd > 0.f && ndz < 0.f) {
      float invd = 1.f / dd;
      sco = 0.f;
      for (int n = 0; n < NVIEW; ++n) {
        float p0 = cbL[n*9+0], p1 = cbL[n*9+1], p2 = cbL[n*9+2];
        float p3 = cbL[n*9+3], p4 = cbL[n*9+4], p5 = cbL[n*9+5];
        float p6 = cbL[n*9+6], p7 = cbL[n*9+7], p8 = cbL[n*9+8];
        float t0 = cbL[36+n*3+0], t1 = cbL[36+n*3+1], t2 = cbL[36+n*3+2];
        float k0 = cbL[48+n*9+0], k1 = cbL[48+n*9+1], k2 = cbL[48+n*9+2];
        float k3 = cbL[48+n*9+3], k4 = cbL[48+n*9+4], k5 = cbL[48+n*9+5];
        float k6 = cbL[48+n*9+6], k7 = cbL[48+n*9+7], k8 = cbL[48+n*9+8];
        const float* dn = dst + n*HW;

        float s1 = 0.f, s2 = 0.f, s3 = 0.f;
        bool ok = true;
        for (int ti = 0; ti < 5 && ok; ++ti) {
          int py = y + ti - 2;
          for (int tj = 0; tj < 5; ++tj) {
            int px = x + tj - 2;
            if (px < 0 || px >= WW || py < 0 || py >= HH) { ok = false; break; }
            float fx = (float)px, fy = (float)py;
            // g = Kinv * [x,y,1]
            float gx = k0*fx + k1*fy + k2;
            float gy = k3*fx + k4*fy + k5;
            float gz = k6*fx + k7*fy + k8;
            // warped = PP*g - PT * (n . g)/d
            float q  = (ndx*gx + ndy*gy + ndz*gz) * invd;
            float wx = p0*gx + p1*gy + p2*gz - t0*q;
            float wy = p3*gx + p4*gy + p5*gz - t1*q;
            float wz = p6*gx + p7*gy + p8*gz - t2*q;
            if (!(wz > 0.f)) { ok = false; break; }
            float xw = wx / wz, yw = wy / wz;
            if (!(xw > 0.f && xw < (float)(WW-1) && yw > 0.f && yw < (float)(HH-1))) {
              ok = false; break;
            }
            float v = dn[(int)yw * WW + (int)xw];   // nearest-neighbor gather
            s1 += v;
            s2 += v*v;
            s3 += scL[(ti*5 + tj)*256 + tid] * v;
          }
        }
        float nccv = THRESH;
        if (ok) {
          manyd = 1;
          float dmean = s1 * (1.f/25.f);
          float dvar  = s2 * (1.f/25.f) - dmean*dmean;
          float prod  = svar * dvar;
          if (prod > 0.f)
            nccv = (s3 * (1.f/25.f) - dmean*scm) / sqrtf(prod);
        }
        sco += nccv;
      }
      sco *= 0.25f;                 // mean over views
      if (!manyd) sco = THRESH;     // many = any(mall) & (d>0)
    } else {
      sco = THRESH;
    }
    if (sco > best) { best = sco; bidx = d; bmany = manyd; }   // first-max argmax
  }

  out[0*HW + pix] = 1.f - best;
  out[1*HW + pix] = dpL[bidx*256 + tid];
  out[2*HW + pix] = npL[(bidx*3 + 0)*256 + tid];
  out[3*HW + pix] = npL[(bidx*3 + 1)*256 + tid];
  out[4*HW + pix] = npL[(bidx*3 + 2)*256 + tid];
  out[5*HW + pix] = bmany ? 1.f : 0.f;
}

extern "C" void kernel_launch(void* const* d_in, const int* in_sizes, int n_in,
                              void* d_out, int out_size, void* d_ws, size_t ws_size,
                              hipStream_t stream) {
  (void)in_sizes; (void)n_in; (void)out_size; (void)d_ws; (void)ws_size;
  const float* src   = (const float*)d_in[0];
  const float* dst   = (const float*)d_in[1];
  const float* Ks    = (const float*)d_in[2];
  const float* Rs    = (const float*)d_in[3];
  const float* Ts    = (const float*)d_in[4];
  const float* depth = (const float*)d_in[5];
  const float* norm  = (const float*)d_in[6];
  const int*   dil   = (const int*)d_in[7];
  float* out = (float*)d_out;

  dim3 grid((HW + 255) / 256);
  ncc_main<<<grid, 256, 0, stream>>>(src, dst, Ks, Rs, Ts, depth, norm, dil, out);
}